// VirtualNodePyg_65128884076584
// MI455X (gfx1250) — compile-verified
//
#include <hip/hip_runtime.h>
#include <hip/hip_bf16.h>
#include <math.h>

typedef __attribute__((ext_vector_type(2))) float v2f;
typedef __attribute__((ext_vector_type(8))) float v8f;

// ---------------------------------------------------------------------------
// Stage 1: segmented logsum pooling. batch[] is sorted, so each graph's node
// range is found with two binary searches; no atomics required.
// X[b,:] = (seg_sum / cnt) * log(cnt) + vn_feat[b,:]
// ---------------------------------------------------------------------------
__global__ void pool_kernel(const float* __restrict__ feat,
                            const float* __restrict__ vn,
                            const int* __restrict__ batch,
                            float* __restrict__ X,
                            int N, int D) {
  const int b = blockIdx.x;

  // lower_bound(batch, b)
  int lo = 0, hi = N;
  while (lo < hi) { int mid = (lo + hi) >> 1; if (batch[mid] < b) lo = mid + 1; else hi = mid; }
  const int start = lo;
  // lower_bound(batch, b+1)
  hi = N;
  while (lo < hi) { int mid = (lo + hi) >> 1; if (batch[mid] < b + 1) lo = mid + 1; else hi = mid; }
  const int end = lo;

  const float cnt = (float)(end - start);
  const float scale = (cnt > 0.0f) ? (__logf(cnt) / cnt) : 0.0f;

  for (int d = threadIdx.x; d < D; d += blockDim.x) {
    float s = 0.0f;
    for (int r = start; r < end; ++r)
      s += feat[(size_t)r * D + d];   // consecutive threads -> consecutive addrs: coalesced
    X[(size_t)b * D + d] = s * scale + vn[(size_t)b * D + d];
  }
}

// ---------------------------------------------------------------------------
// Stage 2: fp32 GEMM on the matrix pipe: V_WMMA_F32_16X16X4_F32.
// One 32-lane wave owns a 32x32 output macro-tile (2 M-tiles x 2 N-tiles,
// 4 independent accumulators). Per K-step: 4 operand loads feed 4 WMMAs
// (1.0 load/WMMA) and the 4 chains are independent, hiding the WMMA->WMMA
// RAW latency from the ISA hazard table.
// epilogue==1 : out = add[] + relu(acc + bias)      (fc layer + residual)
// epilogue==0 : out = acc + bias                    (projection)
// Block = 32 threads, straight-line wave code -> EXEC all ones (ISA req).
// ---------------------------------------------------------------------------
__global__ void wmma_gemm(const float* __restrict__ A,     // [M,K] row-major
                          const float* __restrict__ W,     // [K,Nc] row-major
                          const float* __restrict__ bias,  // [Nc]
                          const float* __restrict__ add,   // [M,Nc] or ignored
                          float* __restrict__ out,         // [M,Nc]
                          int K, int Nc, int epilogue) {
  const int lane  = threadIdx.x;        // 0..31
  const int half  = lane >> 4;          // 0 | 1
  const int mlane = lane & 15;          // 0..15
  const int tileN = blockIdx.x * 32;
  const int tileM = blockIdx.y * 32;

  // A 16x4 f32 layout: lane l holds A[l&15, k0 + (l>>4)*2 + {0,1}]
  // B 4x16 f32 layout: lane l holds B[k0 + (l>>4)*2 + {0,1}, l&15]
  const float* a0p = A + (size_t)(tileM + mlane) * K + half * 2;        // M tile 0
  const float* a1p = a0p + (size_t)16 * K;                              // M tile 1
  const float* b0p = W + (size_t)(half * 2) * Nc + tileN + mlane;       // N tile 0
  const float* b1p = b0p + 16;                                          // N tile 1

  v8f acc00 = {}, acc01 = {}, acc10 = {}, acc11 = {};

  #pragma unroll 2
  for (int k = 0; k < K; k += 4) {
    v2f a0, a1, b0, b1;
    a0.x = a0p[k];  a0.y = a0p[k + 1];
    a1.x = a1p[k];  a1.y = a1p[k + 1];
    const size_t ko = (size_t)k * Nc;
    b0.x = b0p[ko]; b0.y = b0p[ko + Nc];
    b1.x = b1p[ko]; b1.y = b1p[ko + Nc];
    // (neg_a, A, neg_b, B, c_mod, C, reuse_a, reuse_b)
    acc00 = __builtin_amdgcn_wmma_f32_16x16x4_f32(false, a0, false, b0,
                                                  (short)0, acc00, false, false);
    acc01 = __builtin_amdgcn_wmma_f32_16x16x4_f32(false, a0, false, b1,
                                                  (short)0, acc01, false, false);
    acc10 = __builtin_amdgcn_wmma_f32_16x16x4_f32(false, a1, false, b0,
                                                  (short)0, acc10, false, false);
    acc11 = __builtin_amdgcn_wmma_f32_16x16x4_f32(false, a1, false, b1,
                                                  (short)0, acc11, false, false);
  }

  // C/D layout: element v of lane l -> row (v + 8*(l>>4)), col (l&15)
  const int col0 = tileN + mlane;
  const int col1 = col0 + 16;
  const float bv0 = bias[col0];
  const float bv1 = bias[col1];

  #pragma unroll
  for (int v = 0; v < 8; ++v) {
    const int row0 = tileM + v + half * 8;
    const int row1 = row0 + 16;
    const size_t i00 = (size_t)row0 * Nc + col0;
    const size_t i01 = (size_t)row0 * Nc + col1;
    const size_t i10 = (size_t)row1 * Nc + col0;
    const size_t i11 = (size_t)row1 * Nc + col1;
    float r00 = acc00[v] + bv0;
    float r01 = acc01[v] + bv1;
    float r10 = acc10[v] + bv0;
    float r11 = acc11[v] + bv1;
    if (epilogue) {
      r00 = add[i00] + fmaxf(r00, 0.0f);
      r01 = add[i01] + fmaxf(r01, 0.0f);
      r10 = add[i10] + fmaxf(r10, 0.0f);
      r11 = add[i11] + fmaxf(r11, 0.0f);
    }
    out[i00] = r00;
    out[i01] = r01;
    out[i10] = r10;
    out[i11] = r11;
  }
}

// ---------------------------------------------------------------------------
// Stage 3: feat_out[i,:] = feat[i,:] + P[batch[i],:]   (float4 streamed)
// One block per node; 192 threads cover 768/4 float4 lanes.
// ---------------------------------------------------------------------------
__global__ void scatter_kernel(const float* __restrict__ feat,
                               const float* __restrict__ P,
                               const int* __restrict__ batch,
                               float* __restrict__ out,
                               int D4 /* = D/4 */) {
  const size_t node = blockIdx.x;
  const int b = batch[node];
  const float4* f = (const float4*)feat + node * D4;
  const float4* p = (const float4*)P + (size_t)b * D4;
  float4* o = (float4*)out + node * D4;
  for (int d = threadIdx.x; d < D4; d += blockDim.x) {
    float4 fv = f[d], pv = p[d];
    float4 ov = { fv.x + pv.x, fv.y + pv.y, fv.z + pv.z, fv.w + pv.w };
    o[d] = ov;
  }
}

// ---------------------------------------------------------------------------
extern "C" void kernel_launch(void* const* d_in, const int* in_sizes, int n_in,
                              void* d_out, int out_size, void* d_ws, size_t ws_size,
                              hipStream_t stream) {
  const float* feat    = (const float*)d_in[0];   // [N,D]
  const float* vn_feat = (const float*)d_in[1];   // [B,D]
  const float* edge    = (const float*)d_in[2];   // [B,DE]
  const int*   batch   = (const int*)d_in[3];     // [N]
  const float* fc_W    = (const float*)d_in[5];   // [D,D]
  const float* fc_b    = (const float*)d_in[6];   // [D]
  const float* proj_W  = (const float*)d_in[7];   // [D,D]
  const float* proj_b  = (const float*)d_in[8];   // [D]

  const int N  = in_sizes[3];
  const int D  = in_sizes[6];
  const int B  = in_sizes[1] / D;
  const int ED = in_sizes[2];                     // B*DE elements

  float* feat_out = (float*)d_out;                         // [N,D]
  float* vn_new   = feat_out + (size_t)N * D;              // [B,D]
  float* edge_out = vn_new + (size_t)B * D;                // [B,DE]

  float* X = (float*)d_ws;                                 // [B,D] gemm1 input
  float* P = X + (size_t)B * D;                            // [B,D] projection

  // 1) logsum pooling + vn_feat  ->  X
  pool_kernel<<<B, 256, 0, stream>>>(feat, vn_feat, batch, X, N, D);

  // 2) vn_new = vn_feat + relu(X @ fc_W + fc_b)   (WMMA f32, exact)
  {
    dim3 grid(D / 32, B / 32);
    wmma_gemm<<<grid, 32, 0, stream>>>(X, fc_W, fc_b, vn_feat, vn_new, D, D, 1);
  }

  // 3) P = vn_new @ proj_W + proj_b               (WMMA f32, exact)
  {
    dim3 grid(D / 32, B / 32);
    wmma_gemm<<<grid, 32, 0, stream>>>(vn_new, proj_W, proj_b, nullptr, P, D, D, 0);
  }

  // 4) feat_out = feat + P[batch]                 (bandwidth-bound stream)
  scatter_kernel<<<N, 192, 0, stream>>>(feat, P, batch, feat_out, D / 4);

  // 5) edge features pass through unchanged
  hipMemcpyAsync(edge_out, edge, (size_t)ED * sizeof(float),
                 hipMemcpyDeviceToDevice, stream);
}